// AttentiveBimodalCSRPool_39737037423403
// MI455X (gfx1250) — compile-verified
//
#include <hip/hip_runtime.h>
#include <hip/hip_bf16.h>
#include <math.h>

// ---------------------------------------------------------------------------
// AttentiveBimodalCSRPool for MI455X (gfx1250, wave32).
// Memory-bound workload (~400MB traffic vs ~25 GFLOP) -> keep FP32, use
// V_WMMA_F32_16X16X4_F32 for all GEMMs, recompute activations instead of
// spilling 256MB of [V,64] intermediates to HBM.
// ---------------------------------------------------------------------------

#define NGROUPS 65536
#define VELEMS  524288
#define FMAIN   128
#define FMOD    128
#define FPROJ   32
#define H1DIM   64
#define H2DIM   64
#define SDIM    64
#define K1DIM   97          // x_proj | x-pmin | x-pmax | 1/sqrt(n)
#define BN_EPSF 1e-5f
#define SM_EPSF 1e-12f

typedef __attribute__((ext_vector_type(2))) float v2f;
typedef __attribute__((ext_vector_type(8))) float v8f;

// Order-preserving float<->uint encoding for atomic min/max on floats.
__device__ __forceinline__ unsigned ordered_enc(float f) {
  unsigned u = __float_as_uint(f);
  return (u & 0x80000000u) ? ~u : (u | 0x80000000u);
}
__device__ __forceinline__ float ordered_dec(unsigned u) {
  return (u & 0x80000000u) ? __uint_as_float(u ^ 0x80000000u)
                           : __uint_as_float(~u);
}

// D(16x16,f32) += A(16x4,f32) * B(4x16,f32)
__device__ __forceinline__ v8f wmma_f32(v2f a, v2f b, v8f c) {
  return __builtin_amdgcn_wmma_f32_16x16x4_f32(
      /*neg_a=*/false, a, /*neg_b=*/false, b,
      /*c_mod=*/(short)0, c, /*reuse_a=*/false, /*reuse_b=*/false);
}

// ---------------------------------------------------------------------------
// Init scratch: pmin=+enc-inf, pmax=0, cmax=0, denom=0, BN stats=0.
__global__ void init_ws(unsigned* pminu, unsigned* pmaxu, unsigned* cmaxu,
                        float* denom, float* stats) {
  int i = blockIdx.x * blockDim.x + threadIdx.x;
  if (i < NGROUPS * FPROJ) { pminu[i] = 0xFFFFFFFFu; pmaxu[i] = 0u; }
  if (i < NGROUPS)         { cmaxu[i] = 0u; denom[i] = 0.0f; }
  if (i < 512)             { stats[i] = 0.0f; }
}

// seg[v] = searchsorted(csr, v, 'right') - 1  (largest g with csr[g] <= v)
__global__ void seg_kernel(const long long* __restrict__ csr,
                           int* __restrict__ seg) {
  int v = blockIdx.x * blockDim.x + threadIdx.x;
  if (v >= VELEMS) return;
  int lo = 0, hi = NGROUPS - 1;
  while (lo < hi) {
    int mid = (lo + hi + 1) >> 1;
    if (csr[mid] <= (long long)v) lo = mid; else hi = mid - 1;
  }
  seg[v] = lo;
}

__global__ void group_kernel(const long long* __restrict__ csr,
                             float* __restrict__ invs) {
  int g = blockIdx.x * blockDim.x + threadIdx.x;
  if (g >= NGROUPS) return;
  long long c = csr[g + 1] - csr[g];
  invs[g] = rsqrtf((float)(c > 0 ? c : 1));
}

// Per-(group,feature) min/max of x_proj via ordered-uint atomics.
__global__ void minmax_kernel(const float* __restrict__ xp,
                              const int* __restrict__ seg,
                              unsigned* __restrict__ pminu,
                              unsigned* __restrict__ pmaxu) {
  int i = blockIdx.x * blockDim.x + threadIdx.x;
  if (i >= VELEMS * FPROJ) return;
  int v = i >> 5, f = i & 31;
  unsigned u = ordered_enc(xp[i]);
  int g = seg[v];
  atomicMin(&pminu[g * FPROJ + f], u);
  atomicMax(&pmaxu[g * FPROJ + f], u);
}

// ---------------------------------------------------------------------------
// Qmat[N,64] = x_main[N,128] @ Wq[128,64] + bq.  One wave per 16x16 tile,
// 4 waves/block (one per column tile), Wq staged in LDS.
__global__ void __launch_bounds__(128)
q_gemm(const float* __restrict__ xmain, const float* __restrict__ Wq,
       const float* __restrict__ bq, float* __restrict__ Qmat) {
  __shared__ float wlds[FMAIN * SDIM];  // 32KB
  for (int i = threadIdx.x; i < FMAIN * SDIM; i += 128) wlds[i] = Wq[i];
  __syncthreads();

  int wave = threadIdx.x >> 5, lane = threadIdx.x & 31;
  int col0 = lane & 15;
  int koff = (lane >> 4) << 1;      // lanes 16..31 hold K=2,3 of the A frag
  int roff = (lane >> 4) << 3;      // lanes 16..31 hold rows 8..15 of C/D
  int m0 = blockIdx.x * 16;
  int col = wave * 16 + col0;

  v8f acc;
  {
    float bv = bq[col];
#pragma unroll
    for (int r = 0; r < 8; ++r) acc[r] = bv;
  }
  const float* arow = xmain + (size_t)(m0 + col0) * FMAIN;
  for (int k = 0; k < FMAIN; k += 4) {
    v2f a; a.x = arow[k + koff]; a.y = arow[k + koff + 1];
    v2f b; b.x = wlds[(k + koff) * SDIM + col];
           b.y = wlds[(k + koff + 1) * SDIM + col];
    acc = wmma_f32(a, b, acc);
  }
#pragma unroll
  for (int r = 0; r < 8; ++r)
    Qmat[(size_t)(m0 + r + roff) * SDIM + col] = acc[r];
}

// ---------------------------------------------------------------------------
// feat column k of row `row` (group g), zero-padded K 97->100.
__device__ __forceinline__ float featval(int k, const float* __restrict__ xpr,
                                         int g,
                                         const unsigned* __restrict__ pminu,
                                         const unsigned* __restrict__ pmaxu,
                                         const float* __restrict__ invs) {
  if (k < FPROJ)     return xpr[k];
  if (k < 2 * FPROJ) return xpr[k - FPROJ] - ordered_dec(pminu[g * FPROJ + (k - FPROJ)]);
  if (k < 3 * FPROJ) return xpr[k - 2 * FPROJ] - ordered_dec(pmaxu[g * FPROJ + (k - 2 * FPROJ)]);
  if (k == 96)       return invs[g];
  return 0.0f;
}

// Stage 1 GEMM for one 16-row tile: acc[t] = feat(16x97) @ W1(:,t*16..) + b1.
__device__ __forceinline__ void mlp_stage1(int v0, int lane,
    const float* __restrict__ xp, const int* __restrict__ seg,
    const unsigned* __restrict__ pminu, const unsigned* __restrict__ pmaxu,
    const float* __restrict__ invs, const float* __restrict__ W1,
    const float* __restrict__ b1v, v8f acc[4]) {
  int col0 = lane & 15;
  int koff = (lane >> 4) << 1;
  int row = v0 + col0;
  int g = seg[row];
  const float* xpr = xp + (size_t)row * FPROJ;
#pragma unroll
  for (int t = 0; t < 4; ++t) {
    float bv = b1v[t * 16 + col0];
#pragma unroll
    for (int r = 0; r < 8; ++r) acc[t][r] = bv;
  }
  for (int k = 0; k < 100; k += 4) {  // 25 K-steps, tail zero-padded
    v2f a;
    a.x = featval(k + koff,     xpr, g, pminu, pmaxu, invs);
    a.y = featval(k + koff + 1, xpr, g, pminu, pmaxu, invs);
#pragma unroll
    for (int t = 0; t < 4; ++t) {
      int col = t * 16 + col0;
      v2f b;
      b.x = (k + koff     < K1DIM) ? W1[(k + koff) * H1DIM + col]     : 0.0f;
      b.y = (k + koff + 1 < K1DIM) ? W1[(k + koff + 1) * H1DIM + col] : 0.0f;
      acc[t] = wmma_f32(a, b, acc[t]);
    }
  }
}

// Stage 2: BN1+ReLU(acc1) staged through per-wave LDS tile, then @W2 + b2.
__device__ __forceinline__ void mlp_stage2(int lane, float* __restrict__ hbuf,
    const v8f acc1[4], const float* __restrict__ a1, const float* __restrict__ b1n,
    const float* __restrict__ W2, const float* __restrict__ b2v, v8f acc2[4]) {
  int col0 = lane & 15;
  int koff = (lane >> 4) << 1;
  int roff = (lane >> 4) << 3;
#pragma unroll
  for (int t = 0; t < 4; ++t) {
    int col = t * 16 + col0;
    float aa = a1[col], bb = b1n[col];
#pragma unroll
    for (int r = 0; r < 8; ++r)
      hbuf[(r + roff) * H1DIM + col] = fmaxf(aa * acc1[t][r] + bb, 0.0f);
  }
  __syncthreads();  // C-layout -> A-layout relayout through LDS
#pragma unroll
  for (int t = 0; t < 4; ++t) {
    float bv = b2v[t * 16 + col0];
#pragma unroll
    for (int r = 0; r < 8; ++r) acc2[t][r] = bv;
  }
  for (int k = 0; k < H1DIM; k += 4) {
    v2f a; a.x = hbuf[col0 * H1DIM + k + koff];
           a.y = hbuf[col0 * H1DIM + k + koff + 1];
#pragma unroll
    for (int t = 0; t < 4; ++t) {
      int col = t * 16 + col0;
      v2f b; b.x = W2[(k + koff) * H2DIM + col];
             b.y = W2[(k + koff + 1) * H2DIM + col];
      acc2[t] = wmma_f32(a, b, acc2[t]);
    }
  }
}

// ---------------------------------------------------------------------------
// Pass A: accumulate BatchNorm1 batch statistics of h1 = feat@W1 + b1.
__global__ void __launch_bounds__(64)
stats1_kernel(const float* __restrict__ xp, const int* __restrict__ seg,
              const unsigned* __restrict__ pminu, const unsigned* __restrict__ pmaxu,
              const float* __restrict__ invs, const float* __restrict__ W1,
              const float* __restrict__ b1v, float* __restrict__ sum1,
              float* __restrict__ sumsq1) {
  int wave = threadIdx.x >> 5, lane = threadIdx.x & 31;
  int v0 = (blockIdx.x * 2 + wave) * 16;
  v8f acc[4];
  mlp_stage1(v0, lane, xp, seg, pminu, pmaxu, invs, W1, b1v, acc);
  int col0 = lane & 15;
#pragma unroll
  for (int t = 0; t < 4; ++t) {
    float s = 0.0f, ss = 0.0f;
#pragma unroll
    for (int r = 0; r < 8; ++r) { float x = acc[t][r]; s += x; ss += x * x; }
    atomicAdd(&sum1[t * 16 + col0], s);
    atomicAdd(&sumsq1[t * 16 + col0], ss);
  }
}

// stats -> BN affine: y = a*x + b  with a = gamma*rsqrt(var+eps), b = beta - m*a
__global__ void finalize_bn(const float* __restrict__ sum,
                            const float* __restrict__ sumsq,
                            const float* __restrict__ gamma,
                            const float* __restrict__ beta,
                            float* __restrict__ a_out, float* __restrict__ b_out) {
  int c = threadIdx.x;
  if (c >= 64) return;
  float m = sum[c] / (float)VELEMS;
  float var = sumsq[c] / (float)VELEMS - m * m;
  float a = gamma[c] * rsqrtf(var + BN_EPSF);
  a_out[c] = a;
  b_out[c] = beta[c] - m * a;
}

// Pass B: recompute h1, BN1+ReLU, h2 = @W2 + b2; accumulate BN2 stats.
__global__ void __launch_bounds__(64)
stats2_kernel(const float* __restrict__ xp, const int* __restrict__ seg,
              const unsigned* __restrict__ pminu, const unsigned* __restrict__ pmaxu,
              const float* __restrict__ invs, const float* __restrict__ W1,
              const float* __restrict__ b1v, const float* __restrict__ a1,
              const float* __restrict__ b1n, const float* __restrict__ W2,
              const float* __restrict__ b2v, float* __restrict__ sum2,
              float* __restrict__ sumsq2) {
  __shared__ float hbuf[2][16 * H1DIM];
  int wave = threadIdx.x >> 5, lane = threadIdx.x & 31;
  int v0 = (blockIdx.x * 2 + wave) * 16;
  v8f acc1[4], acc2[4];
  mlp_stage1(v0, lane, xp, seg, pminu, pmaxu, invs, W1, b1v, acc1);
  mlp_stage2(lane, hbuf[wave], acc1, a1, b1n, W2, b2v, acc2);
  int col0 = lane & 15;
#pragma unroll
  for (int t = 0; t < 4; ++t) {
    float s = 0.0f, ss = 0.0f;
#pragma unroll
    for (int r = 0; r < 8; ++r) { float x = acc2[t][r]; s += x; ss += x * x; }
    atomicAdd(&sum2[t * 16 + col0], s);
    atomicAdd(&sumsq2[t * 16 + col0], ss);
  }
}

// Pass C: full MLP -> K = [h, x_mod] @ Wk + bk -> C = <K, Q[seg]> / 8.
__global__ void __launch_bounds__(64)
score_kernel(const float* __restrict__ xp, const int* __restrict__ seg,
             const unsigned* __restrict__ pminu, const unsigned* __restrict__ pmaxu,
             const float* __restrict__ invs, const float* __restrict__ W1,
             const float* __restrict__ b1v, const float* __restrict__ a1,
             const float* __restrict__ b1n, const float* __restrict__ W2,
             const float* __restrict__ b2v, const float* __restrict__ a2,
             const float* __restrict__ b2n, const float* __restrict__ xmod,
             const float* __restrict__ Wk, const float* __restrict__ bkv,
             const float* __restrict__ Qmat, float* __restrict__ Cbuf,
             unsigned* __restrict__ cmaxu) {
  __shared__ float hbuf[2][16 * H1DIM];
  __shared__ float cpart[2][16];
  int wave = threadIdx.x >> 5, lane = threadIdx.x & 31;
  int v0 = (blockIdx.x * 2 + wave) * 16;
  int col0 = lane & 15;
  int koff = (lane >> 4) << 1;
  int roff = (lane >> 4) << 3;

  v8f acc1[4], acc2[4], acc3[4];
  mlp_stage1(v0, lane, xp, seg, pminu, pmaxu, invs, W1, b1v, acc1);
  mlp_stage2(lane, hbuf[wave], acc1, a1, b1n, W2, b2v, acc2);
  __syncthreads();
  // BN2 + ReLU, write h back into the LDS tile
#pragma unroll
  for (int t = 0; t < 4; ++t) {
    int col = t * 16 + col0;
    float aa = a2[col], bb = b2n[col];
#pragma unroll
    for (int r = 0; r < 8; ++r)
      hbuf[wave][(r + roff) * H1DIM + col] = fmaxf(aa * acc2[t][r] + bb, 0.0f);
  }
  __syncthreads();

  // K tile = h(16x64)@Wk[0:64] + xmod(16x128)@Wk[64:192] + bk
#pragma unroll
  for (int t = 0; t < 4; ++t) {
    float bv = bkv[t * 16 + col0];
#pragma unroll
    for (int r = 0; r < 8; ++r) acc3[t][r] = bv;
  }
  for (int k = 0; k < H2DIM; k += 4) {
    v2f a; a.x = hbuf[wave][col0 * H1DIM + k + koff];
           a.y = hbuf[wave][col0 * H1DIM + k + koff + 1];
#pragma unroll
    for (int t = 0; t < 4; ++t) {
      int col = t * 16 + col0;
      v2f b; b.x = Wk[(k + koff) * SDIM + col];
             b.y = Wk[(k + koff + 1) * SDIM + col];
      acc3[t] = wmma_f32(a, b, acc3[t]);
    }
  }
  const float* xmr = xmod + (size_t)(v0 + col0) * FMOD;
  for (int k = 0; k < FMOD; k += 4) {
    __builtin_prefetch(xmr + k + 32, 0, 0);
    v2f a; a.x = xmr[k + koff]; a.y = xmr[k + koff + 1];
#pragma unroll
    for (int t = 0; t < 4; ++t) {
      int col = t * 16 + col0;
      v2f b; b.x = Wk[(H2DIM + k + koff) * SDIM + col];
             b.y = Wk[(H2DIM + k + koff + 1) * SDIM + col];
      acc3[t] = wmma_f32(a, b, acc3[t]);
    }
  }

  // C[v] = sum_s K[v][s] * Q[seg[v]][s] / sqrt(S); reduce across lanes via LDS.
  if (lane < 16) cpart[wave][lane] = 0.0f;
#pragma unroll
  for (int r = 0; r < 8; ++r) {
    int m = r + roff;
    int g = seg[v0 + m];
    float s = 0.0f;
#pragma unroll
    for (int t = 0; t < 4; ++t)
      s += acc3[t][r] * Qmat[(size_t)g * SDIM + t * 16 + col0];
    atomicAdd(&cpart[wave][m], s);
  }
  __syncthreads();
  if (lane < 16) {
    int v = v0 + lane;
    float c = cpart[wave][lane] * 0.125f;  // 1/sqrt(64)
    Cbuf[v] = c;
    atomicMax(&cmaxu[seg[v]], ordered_enc(c));
  }
}

// e = exp(C - cmax[g]); denom[g] += e   (Cbuf overwritten in place with e)
__global__ void expsum_kernel(const int* __restrict__ seg,
                              const unsigned* __restrict__ cmaxu,
                              float* __restrict__ Cbuf,
                              float* __restrict__ denom) {
  int v = blockIdx.x * blockDim.x + threadIdx.x;
  if (v >= VELEMS) return;
  int g = seg[v];
  float e = expf(Cbuf[v] - ordered_dec(cmaxu[g]));
  Cbuf[v] = e;
  atomicAdd(&denom[g], e);
}

// One wave per group: x_pool = sum_v x_mod[v]*A[v], gated; write x_pool + seen.
__global__ void __launch_bounds__(128)
pool_kernel(const long long* __restrict__ csr, const float* __restrict__ xmod,
            const float* __restrict__ Ebuf, const float* __restrict__ denom,
            const unsigned* __restrict__ cmaxu, const float* __restrict__ gw,
            const float* __restrict__ gb, float* __restrict__ out) {
  int wv = (blockIdx.x * blockDim.x + threadIdx.x) >> 5;
  int lane = threadIdx.x & 31;
  if (wv >= NGROUPS) return;
  long long s = csr[wv], e = csr[wv + 1];
  float inv = 1.0f / (denom[wv] + SM_EPSF);
  float a0 = 0.f, a1 = 0.f, a2 = 0.f, a3 = 0.f;
  for (long long v = s; v < e; ++v) {
    float a = Ebuf[v] * inv;
    const float* xr = xmod + (size_t)v * FMOD;
    __builtin_prefetch(xr + FMOD, 0, 0);
    a0 += xr[lane] * a;       a1 += xr[lane + 32] * a;
    a2 += xr[lane + 64] * a;  a3 += xr[lane + 96] * a;
  }
  float cm = (e > s) ? ordered_dec(cmaxu[wv]) : 0.0f;
  float G = tanhf(fmaxf(gw[0] * cm + gb[0], 0.0f));
  float* o = out + (size_t)wv * FMOD;
  o[lane] = a0 * G;      o[lane + 32] = a1 * G;
  o[lane + 64] = a2 * G; o[lane + 96] = a3 * G;
  if (lane == 0) out[(size_t)NGROUPS * FMOD + wv] = (e > s) ? 1.0f : 0.0f;
}

// ---------------------------------------------------------------------------
extern "C" void kernel_launch(void* const* d_in, const int* in_sizes, int n_in,
                              void* d_out, int out_size, void* d_ws, size_t ws_size,
                              hipStream_t stream) {
  const float*     xmain  = (const float*)d_in[0];
  const float*     xmod   = (const float*)d_in[1];
  const float*     xproj  = (const float*)d_in[2];
  const long long* csr    = (const long long*)d_in[3];
  const float* Wq = (const float*)d_in[4];  const float* bq = (const float*)d_in[5];
  const float* W1 = (const float*)d_in[6];  const float* b1 = (const float*)d_in[7];
  const float* g1 = (const float*)d_in[8];  const float* be1 = (const float*)d_in[9];
  const float* W2 = (const float*)d_in[10]; const float* b2 = (const float*)d_in[11];
  const float* g2 = (const float*)d_in[12]; const float* be2 = (const float*)d_in[13];
  const float* Wk = (const float*)d_in[14]; const float* bk = (const float*)d_in[15];
  const float* gw = (const float*)d_in[16]; const float* gb = (const float*)d_in[17];

  char* ws = (char*)d_ws;
  size_t off = 0;
  auto alloc = [&](size_t bytes) -> void* {
    void* p = ws + off;
    off = (off + bytes + 255) & ~(size_t)255;
    return p;
  };
  int*      seg   = (int*)alloc(4ull * VELEMS);
  float*    invs  = (float*)alloc(4ull * NGROUPS);
  unsigned* pminu = (unsigned*)alloc(4ull * NGROUPS * FPROJ);
  unsigned* pmaxu = (unsigned*)alloc(4ull * NGROUPS * FPROJ);
  float*    Qmat  = (float*)alloc(4ull * NGROUPS * SDIM);
  float*    Cbuf  = (float*)alloc(4ull * VELEMS);
  unsigned* cmaxu = (unsigned*)alloc(4ull * NGROUPS);
  float*    denom = (float*)alloc(4ull * NGROUPS);
  float*    stats = (float*)alloc(4ull * 512);
  float *sum1 = stats,       *sq1 = stats + 64,  *sum2 = stats + 128, *sq2 = stats + 192;
  float *a1   = stats + 256, *b1n = stats + 320, *a2   = stats + 384, *b2n = stats + 448;

  init_ws<<<(NGROUPS * FPROJ + 255) / 256, 256, 0, stream>>>(pminu, pmaxu, cmaxu, denom, stats);
  seg_kernel<<<(VELEMS + 255) / 256, 256, 0, stream>>>(csr, seg);
  group_kernel<<<(NGROUPS + 255) / 256, 256, 0, stream>>>(csr, invs);
  minmax_kernel<<<(VELEMS * FPROJ + 255) / 256, 256, 0, stream>>>(xproj, seg, pminu, pmaxu);
  q_gemm<<<NGROUPS / 16, 128, 0, stream>>>(xmain, Wq, bq, Qmat);
  stats1_kernel<<<VELEMS / 32, 64, 0, stream>>>(xproj, seg, pminu, pmaxu, invs, W1, b1, sum1, sq1);
  finalize_bn<<<1, 64, 0, stream>>>(sum1, sq1, g1, be1, a1, b1n);
  stats2_kernel<<<VELEMS / 32, 64, 0, stream>>>(xproj, seg, pminu, pmaxu, invs, W1, b1,
                                                a1, b1n, W2, b2, sum2, sq2);
  finalize_bn<<<1, 64, 0, stream>>>(sum2, sq2, g2, be2, a2, b2n);
  score_kernel<<<VELEMS / 32, 64, 0, stream>>>(xproj, seg, pminu, pmaxu, invs, W1, b1,
                                               a1, b1n, W2, b2, a2, b2n,
                                               xmod, Wk, bk, Qmat, Cbuf, cmaxu);
  expsum_kernel<<<(VELEMS + 255) / 256, 256, 0, stream>>>(seg, cmaxu, Cbuf, denom);
  pool_kernel<<<(NGROUPS * 32 + 127) / 128, 128, 0, stream>>>(csr, xmod, Cbuf, denom,
                                                              cmaxu, gw, gb, (float*)d_out);
}